// ConditionalPoissonSampler_27754078666940
// MI455X (gfx1250) — compile-verified
//
#include <hip/hip_runtime.h>
#include <cstdint>

// Problem constants (match the reference).
#define DD    2048
#define KK    512
#define NSAMP 32768
#define CSTR  528   // cache row stride in floats (2112 B, 64B-aligned)
#define LROW  516   // LDS row stride in floats

typedef uint32_t u32;
typedef u32 u32x4 __attribute__((ext_vector_type(4)));
typedef int i32x8 __attribute__((ext_vector_type(8)));

__device__ __forceinline__ float logaddexpf_(float a, float b) {
  float m = fmaxf(a, b);
  if (m == -__builtin_inff()) return -__builtin_inff();
  float d = fabsf(a - b);          // one -inf operand -> d=+inf -> exp(-inf)=0
  return m + log1pf(__expf(-d));
}

// ---------------------------------------------------------------------------
// Kernel 1: log-semiring DP.  cache[i][n] = log S_{i+1}(n), n = 0..K.
// One 512-thread workgroup, thread t owns column t+1 (column 0 is const 0).
// Double-buffered LDS row -> exactly one barrier per DP row.
// ---------------------------------------------------------------------------
__global__ __launch_bounds__(512) void build_cache_kernel(
    const float* __restrict__ w, float* __restrict__ cache) {
  __shared__ float buf[2][KK + 2];
  const int tid = threadIdx.x;
  const float NEG_INF = -__builtin_inff();

  float c0 = (tid == 0) ? w[0] : NEG_INF;   // row 0: col1 = w[0], cols>=2 = -inf
  buf[0][tid + 1] = c0;
  if (tid == 0) buf[0][0] = 0.0f;
  cache[tid + 1] = c0;
  if (tid == 0) cache[0] = 0.0f;
  __syncthreads();

  int pb = 0;
  for (int i = 1; i < DD; ++i) {
    const float wi  = w[i];                 // uniform -> scalar load
    const float pm1 = buf[pb][tid];         // prev[n-1]
    const float pv  = buf[pb][tid + 1];     // prev[n]
    const float c   = logaddexpf_(wi + pm1, pv);
    buf[pb ^ 1][tid + 1] = c;
    if (tid == 0) buf[pb ^ 1][0] = 0.0f;    // column 0 stays log S(0)=0
    float* row = cache + (size_t)i * CSTR;
    row[tid + 1] = c;
    if (tid == 0) row[0] = 0.0f;
    __syncthreads();
    pb ^= 1;
  }
}

// ---------------------------------------------------------------------------
// threefry-2x32, 20 rounds, key = jax key(1) -> (0,1).  Counter = (step, sample).
// ---------------------------------------------------------------------------
__device__ __forceinline__ u32 rotl32_(u32 x, int r) { return (x << r) | (x >> (32 - r)); }

__device__ __forceinline__ u32 threefry2x32_(u32 c0, u32 c1) {
  const u32 k0 = 0u, k1 = 1u;
  const u32 k2 = k0 ^ k1 ^ 0x1BD11BDAu;
  u32 x0 = c0 + k0, x1 = c1 + k1;
#define TFR(r) { x0 += x1; x1 = rotl32_(x1, (r)); x1 ^= x0; }
  TFR(13) TFR(15) TFR(26) TFR(6)   x0 += k1; x1 += k2 + 1u;
  TFR(17) TFR(29) TFR(16) TFR(24)  x0 += k2; x1 += k0 + 2u;
  TFR(13) TFR(15) TFR(26) TFR(6)   x0 += k0; x1 += k1 + 3u;
  TFR(17) TFR(29) TFR(16) TFR(24)  x0 += k1; x1 += k2 + 4u;
  TFR(13) TFR(15) TFR(26) TFR(6)   x0 += k2; x1 += k0 + 5u;
#undef TFR
  return x0 ^ x1;
}

// Issue one TDM transfer: cache row `row` -> LDS ring buffer rows[row & 3].
// 2D descriptor, tile 513 x 1, f32.  Wave-0-only call sites.
__device__ __forceinline__ void tdm_prefetch_row(
    int row, const float* __restrict__ cache, float (*rows)[LROW]) {
  const uint64_t ga = (uint64_t)(uintptr_t)(cache + (size_t)row * CSTR);
  const u32 lds = (u32)(uintptr_t)&rows[row & 3][0];
  u32x4 g0;
  g0[0] = 1u;                               // count=1, user descriptor
  g0[1] = lds;                              // lds_addr
  g0[2] = (u32)ga;                          // global_addr[31:0]
  g0[3] = (u32)(ga >> 32) | (2u << 30);     // global_addr[56:32] | type=2
  i32x8 g1;
  g1[0] = 0x00020000;                       // data_size=4B, no multicast
  g1[1] = (int)((KK + 1) << 16);            // tensor_dim0 = 513
  g1[2] = (int)(1 << 16);                   // tensor_dim1 = 1
  g1[3] = (int)((KK + 1) << 16);            // tile_dim0 = 513
  g1[4] = 1;                                // tile_dim1 = 1
  g1[5] = CSTR;                             // tensor_dim0_stride
  g1[6] = 0;
  g1[7] = 0;
  asm volatile("tensor_load_to_lds %0, %1" :: "s"(g0), "s"(g1) : "memory");
}

// ---------------------------------------------------------------------------
// One sampler step for dimension d (d = DD..1).
// Prefetch pipeline depth 2: at step d, row d-3 is already in flight; we issue
// row d-4, compute from LDS rows d-1/d-2, then wait TENSORcnt<=1 (completes
// row d-3) and barrier.  4-buffer LDS ring, row r lives in rows[r & 3].
// ---------------------------------------------------------------------------
__device__ __forceinline__ float sample_step(
    int d, int tid, int s, int& num_sampled,
    const float* __restrict__ w, const float* __restrict__ cache,
    float (*rows)[LROW]) {
  if (d >= 4 && tid < 32) tdm_prefetch_row(d - 4, cache, rows);

  const int rem = KK - num_sampled;
  const float* ra = rows[((d >= 2) ? (d - 2) : (d - 1)) & 3];
  const float* rb = rows[(d - 1) & 3];
  const int ia = min(max(rem - 1, 0), KK);    // value only matters when rem>0
  const int ib = min(max(rem, 0), KK);
  const float a = ra[ia];                     // ds_load_b32 gather
  const float b = rb[ib];
  float p = __expf(w[d - 1] + a - b);
  p = (rem >= d) ? 1.0f : p;                  // must take all remaining dims
  p = (num_sampled >= KK) ? 0.0f : p;         // already full

  const u32 r = threefry2x32_((u32)(DD - d), (u32)s);
  const float uf = (float)r * 2.3283064365386963e-10f;  // [0,1)
  const int bit = (uf < p) ? 1 : 0;
  num_sampled += bit;

  if (d >= 3) {                               // row d-3 must be valid next step
    if (tid < 32) {
      if (d >= 4) {
        __builtin_amdgcn_s_wait_tensorcnt(1); // one transfer stays in flight
      } else {
        __builtin_amdgcn_s_wait_tensorcnt(0); // drain at d==3
      }
    }
    __syncthreads();
  }
  return (float)bit;
}

// ---------------------------------------------------------------------------
// Kernel 2: sampler. 1 sample per thread, 128 threads/block, 256 blocks.
// Initial fill of rows D-1/D-2 via async global->LDS; row D-3 TDM-issued
// before the loop so the pipeline starts with one transfer in flight.
// Output bits packed 4-at-a-time into float4 stores (row-major (N, D)).
// ---------------------------------------------------------------------------
__global__ __launch_bounds__(128) void sampler_kernel(
    const float* __restrict__ w, const float* __restrict__ cache,
    float* __restrict__ out) {
  __shared__ float rows[4][LROW];
  const int tid = threadIdx.x;
  const int s = blockIdx.x * 128 + tid;

  // Initial fill: rows D-1 and D-2 via CDNA5 async global->LDS loads.
  for (int r = 0; r < 2; ++r) {
    const int row = DD - 1 - r;
    const float* src = cache + (size_t)row * CSTR;
    const int buf = row & 3;
    for (int e = tid; e <= KK; e += 128) {
      const u32 lds = (u32)(uintptr_t)&rows[buf][e];
      asm volatile("global_load_async_to_lds_b32 %0, %1, %2"
                   :: "v"(lds), "v"(e * 4), "s"(src) : "memory");
    }
  }
  // Start the TDM pipeline: row D-3 in flight before the first step.
  if (tid < 32) tdm_prefetch_row(DD - 3, cache, rows);
  asm volatile("s_wait_asynccnt 0x0" ::: "memory");
  __syncthreads();

  int num_sampled = 0;
  float* orow = out + (size_t)s * DD;
  for (int d0 = DD; d0 >= 4; d0 -= 4) {
    float4 ob;
    ob.w = sample_step(d0,     tid, s, num_sampled, w, cache, rows);
    ob.z = sample_step(d0 - 1, tid, s, num_sampled, w, cache, rows);
    ob.y = sample_step(d0 - 2, tid, s, num_sampled, w, cache, rows);
    ob.x = sample_step(d0 - 3, tid, s, num_sampled, w, cache, rows);
    *(float4*)(orow + (d0 - 4)) = ob;         // 16B-aligned, dims d0-4..d0-1
  }
}

// ---------------------------------------------------------------------------
extern "C" void kernel_launch(void* const* d_in, const int* in_sizes, int n_in,
                              void* d_out, int out_size, void* d_ws, size_t ws_size,
                              hipStream_t stream) {
  (void)in_sizes; (void)n_in; (void)out_size; (void)ws_size;
  const float* weights = (const float*)d_in[0];
  // d_in[1] = num_samples scalar on device; fixed to 32768 per the reference.
  float* cache = (float*)d_ws;   // needs 2048 * 528 * 4 = 4,325,376 bytes
  float* out = (float*)d_out;

  build_cache_kernel<<<1, 512, 0, stream>>>(weights, cache);
  sampler_kernel<<<NSAMP / 128, 128, 0, stream>>>(weights, cache, out);
}